// BertLexer_1924145349098
// MI455X (gfx1250) — compile-verified
//
#include <hip/hip_runtime.h>

typedef __attribute__((ext_vector_type(2))) float v2f;
typedef __attribute__((ext_vector_type(8))) float v8f;

#define Bsz 16
#define Wsz 256
#define Ssz 512
#define Hsz 768
#define Lsz 12
#define Esz 256
// output row = Esz + Hsz = 1024 floats per (b,w)

__global__ __launch_bounds__(256)
void bert_lexer_kernel(const int* __restrict__ word_indices,
                       const int* __restrict__ span_start,
                       const int* __restrict__ span_end,
                       const float* __restrict__ W_embed,
                       const float* __restrict__ hidden,
                       float* __restrict__ out)
{
    const int bw   = blockIdx.x;      // 0 .. B*W-1, one block per (b,w)
    const int b    = bw >> 8;         // bw / W (W == 256)
    const int t    = threadIdx.x;     // 0..255 (8 waves of 32)
    const int lane = t & 31;
    const int wave = t >> 5;

    const int idx = word_indices[bw];
    const int s0  = span_start[bw];
    const int s1  = span_end[bw];
    const int len = s1 - s0;                       // 1 or 2
    const float wgt = 1.0f / (float)(Lsz * len);   // mean over layers and span

    float* orow = out + (size_t)bw * (size_t)(Esz + Hsz);

    // ---- word embedding gather: 256 contiguous floats, 1 per thread ----
    orow[t] = W_embed[(size_t)idx * Esz + t];

    // ---- span+layer mean via V_WMMA_F32_16X16X4_F32 ----
    // A (16x4): row M=0 holds uniform weight wgt, all other rows zero.
    // A layout: lanes m and m+16 hold row m (2 VGPRs = the 4 K values).
    v2f a;
    a.x = ((lane & 15) == 0) ? wgt : 0.0f;
    a.y = a.x;

    const int  col = lane & 15;       // B/D column (h within tile)
    const bool hi  = lane >= 16;      // high half-wave takes items 2,3 of chunk

    // hidden flat index: ((l*B + b)*S + s)*H + h
    const size_t layer_stride = (size_t)Bsz * Ssz * Hsz;
    const float* hb = hidden + (size_t)b * Ssz * Hsz;

    // 48 tiles of 16 h-columns; 8 waves -> 6 tiles each
    #pragma unroll 1
    for (int tile = wave; tile < Hsz / 16; tile += 8) {
        const int h = tile * 16 + col;
        v8f acc = {0.f, 0.f, 0.f, 0.f, 0.f, 0.f, 0.f, 0.f};

        for (int s = s0; s < s1; ++s) {
            const float* p = hb + (size_t)s * Hsz + h;
            #pragma unroll
            for (int c = 0; c < 3; ++c) {          // 12 layers in chunks of 4
                const int l0 = 4 * c + (hi ? 2 : 0);
                v2f bm;
                bm.x = p[(size_t)l0 * layer_stride];
                bm.y = p[(size_t)(l0 + 1) * layer_stride];
                // D = A x B + C ; uniform A row 0 makes K-lane mapping
                // irrelevant: D[0,n] = wgt * sum of the 4 items in column n.
                acc = __builtin_amdgcn_wmma_f32_16x16x4_f32(
                        /*neg_a=*/false, a,
                        /*neg_b=*/false, bm,
                        /*c_mod=*/(short)0, acc,
                        /*reuse_a=*/false, /*reuse_b=*/false);
            }
        }

        // D[0,n] lives in VGPR0 (component 0) of lanes 0-15
        if (!hi) orow[Esz + tile * 16 + col] = acc[0];
    }
}

extern "C" void kernel_launch(void* const* d_in, const int* in_sizes, int n_in,
                              void* d_out, int out_size, void* d_ws, size_t ws_size,
                              hipStream_t stream) {
    const int*   wi = (const int*)  d_in[0];   // word_indices [B,W]
    const int*   ss = (const int*)  d_in[1];   // span_start   [B,W]
    const int*   se = (const int*)  d_in[2];   // span_end     [B,W]
    const float* we = (const float*)d_in[3];   // W_embed      [V,E]
    const float* hs = (const float*)d_in[4];   // hidden_states [L,B,S,H]
    float* o = (float*)d_out;                  // [B,W,E+H]

    bert_lexer_kernel<<<Bsz * Wsz, 256, 0, stream>>>(wi, ss, se, we, hs, o);
}